// EnergyModel_57423712747890
// MI455X (gfx1250) — compile-verified
//
#include <hip/hip_runtime.h>

// ---------------- problem constants ----------------
#define WDIM 512
#define HDIM 384
#define HW   (WDIM * HDIM)      // 196608
#define BATCH 8
#define NXTOT (BATCH * 5 * HW)  // 7864320 state elements
#define NPIX  (BATCH * HW)      // 1572864 pixels
#define EPS_C       1e-5f
#define MAX_FLOW_C  80.0f
#define NOISE_STD_C 0.05f
#define GRAD_CLIP_C 0.03f
#define STEP_SZ_C   10.0f
#define NSTEPS      200
#define EBLOCKS     24          // reduction blocks per (quantity, batch)

typedef __attribute__((ext_vector_type(2))) float v2f;
typedef __attribute__((ext_vector_type(8))) float v8f;

// ---------------- deterministic counter-based gaussians ----------------
__device__ __forceinline__ unsigned pcg_hash(unsigned v) {
  unsigned s = v * 747796405u + 2891336453u;
  unsigned w = ((s >> ((s >> 28u) + 4u)) ^ s) * 277803737u;
  return (w >> 22u) ^ w;
}
// one Box-Muller -> two gaussians (halves log/sqrt/trig work)
__device__ __forceinline__ void gauss_pair(unsigned key, unsigned step,
                                           float* g0, float* g1) {
  unsigned h1 = pcg_hash(key ^ (step * 0x9E3779B9u + 0x7F4A7C15u));
  unsigned h2 = pcg_hash(h1 ^ 0x85EBCA6Bu);
  float u1 = (float)(h1 >> 8) * 5.9604645e-08f;
  float u2 = (float)(h2 >> 8) * 5.9604645e-08f;
  float r  = sqrtf(-2.0f * __logf(u1 + 1.0e-12f));
  float sn, cs;
  __sincosf(6.28318530718f * u2, &sn, &cs);
  *g0 = r * cs;
  *g1 = r * sn;
}

__device__ __forceinline__ float clampf(float v, float lo, float hi) {
  return fminf(fmaxf(v, lo), hi);
}

// ---------------- kernels ----------------
// x_noised_0 = clip(init + sigma*N_0, -1, 1)   (2 elements / thread, paired RNG)
__global__ void init_noise_kernel(const float* __restrict__ src, float* __restrict__ dst) {
  int t = blockIdx.x * blockDim.x + threadIdx.x;   // 0 .. NXTOT/2-1
  if (t >= NXTOT / 2) return;
  int e0 = 2 * t, e1 = 2 * t + 1;
  float g0, g1;
  gauss_pair((unsigned)t * 2654435761u, 0u, &g0, &g1);
  dst[e0] = clampf(src[e0] + NOISE_STD_C * g0, -1.0f, 1.0f);
  dst[e1] = clampf(src[e1] + NOISE_STD_C * g1, -1.0f, 1.0f);
}

// One Langevin step, fully fused:
//   u    = clip(xn - 10*clip(grad E(xn), +-0.03), -1, 1)
//   xout = add_noise ? clip(u + sigma*N_{next}, -1, 1) : u
// The stencil always reads ALREADY-NOISED values (xn), so noise never has to be
// recomputed for neighbors -> one kernel node per step, one read + one write of x.
__global__ void step_kernel(const float* __restrict__ xn, const float* __restrict__ in2,
                            const float* __restrict__ lw, float* __restrict__ xout,
                            unsigned next_step, int add_noise) {
  // 64x4 pixel tile per 256-thread block for stencil halo reuse in WGP$/L1
  const int tx = threadIdx.x & 63;
  const int ty = threadIdx.x >> 6;
  const int tileX = blockIdx.x & 7;          // 512/64 = 8 tiles across
  const int tileY = blockIdx.x >> 3;         // 384/4  = 96 tiles down
  const int b = blockIdx.y;
  const int w = tileX * 64 + tx;
  const int h = tileY * 4 + ty;
  const int p = h * WDIM + w;

  const float w0s = __expf(lw[0]) * (1.0f / (float)HW);
  const float w1s = __expf(lw[1]) * (1.0f / (float)HW);

  const float* X  = xn  + b * 5 * HW;
  const float* I2 = in2 + b * 3 * HW;

  // ---- data term gradient (pointwise, channels 2..4) ----
  float m0 = X[2 * HW + p], m1 = X[3 * HW + p], m2 = X[4 * HW + p];
  float d0 = 0.5f * (m0 + 1.0f) - I2[p];
  float d1 = 0.5f * (m1 + 1.0f) - I2[HW + p];
  float d2 = 0.5f * (m2 + 1.0f) - I2[2 * HW + p];
  float invA = rsqrtf(d0 * d0 + d1 * d1 + d2 * d2 + EPS_C);
  float gm0 = w0s * 0.5f * d0 * invA;
  float gm1 = w0s * 0.5f * d1 * invA;
  float gm2 = w0s * 0.5f * d2 * invA;

  // ---- smoothness gradient (3x3 stencil on scaled flow, channels 0..1) ----
  #define FLOW(c, hh, ww) (X[(c) * HW + (hh) * WDIM + (ww)] * MAX_FLOW_C)
  const bool vr = (w < WDIM - 1), vd = (h < HDIM - 1), vl = (w > 0), vu = (h > 0);
  float f0 = FLOW(0, h, w), f1 = FLOW(1, h, w);

  float dxc0 = vr ? FLOW(0, h, w + 1) - f0 : 0.0f;
  float dxc1 = vr ? FLOW(1, h, w + 1) - f1 : 0.0f;
  float dyc0 = vd ? FLOW(0, h + 1, w) - f0 : 0.0f;
  float dyc1 = vd ? FLOW(1, h + 1, w) - f1 : 0.0f;
  float invSc = rsqrtf(dxc0 * dxc0 + dxc1 * dxc1 + dyc0 * dyc0 + dyc1 * dyc1 + EPS_C);
  float g0 = -(dxc0 + dyc0) * invSc;
  float g1 = -(dxc1 + dyc1) * invSc;

  if (vl) {  // S at (h, w-1)
    float l0 = FLOW(0, h, w - 1), l1 = FLOW(1, h, w - 1);
    float dxl0 = f0 - l0, dxl1 = f1 - l1;
    float dyl0 = vd ? FLOW(0, h + 1, w - 1) - l0 : 0.0f;
    float dyl1 = vd ? FLOW(1, h + 1, w - 1) - l1 : 0.0f;
    float invSl = rsqrtf(dxl0 * dxl0 + dxl1 * dxl1 + dyl0 * dyl0 + dyl1 * dyl1 + EPS_C);
    g0 += dxl0 * invSl;
    g1 += dxl1 * invSl;
  }
  if (vu) {  // S at (h-1, w)
    float u0 = FLOW(0, h - 1, w), u1 = FLOW(1, h - 1, w);
    float dyu0 = f0 - u0, dyu1 = f1 - u1;
    float dxu0 = vr ? FLOW(0, h - 1, w + 1) - u0 : 0.0f;
    float dxu1 = vr ? FLOW(1, h - 1, w + 1) - u1 : 0.0f;
    float invSu = rsqrtf(dxu0 * dxu0 + dxu1 * dxu1 + dyu0 * dyu0 + dyu1 * dyu1 + EPS_C);
    g0 += dyu0 * invSu;
    g1 += dyu1 * invSu;
  }
  g0 *= w1s * MAX_FLOW_C;
  g1 *= w1s * MAX_FLOW_C;
  #undef FLOW

  // ---- clipped descent update ----
  float u0 = clampf(X[0*HW+p] - STEP_SZ_C * clampf(g0,  -GRAD_CLIP_C, GRAD_CLIP_C), -1.f, 1.f);
  float u1 = clampf(X[1*HW+p] - STEP_SZ_C * clampf(g1,  -GRAD_CLIP_C, GRAD_CLIP_C), -1.f, 1.f);
  float u2 = clampf(m0        - STEP_SZ_C * clampf(gm0, -GRAD_CLIP_C, GRAD_CLIP_C), -1.f, 1.f);
  float u3 = clampf(m1        - STEP_SZ_C * clampf(gm1, -GRAD_CLIP_C, GRAD_CLIP_C), -1.f, 1.f);
  float u4 = clampf(m2        - STEP_SZ_C * clampf(gm2, -GRAD_CLIP_C, GRAD_CLIP_C), -1.f, 1.f);

  // ---- fused: apply NEXT step's elementwise noise ----
  if (add_noise) {
    const unsigned t = (unsigned)(b * HW + p);
    float n0, n1, n2, n3, n4, nx;
    gauss_pair(t * 3u + 0u, next_step, &n0, &n1);
    gauss_pair(t * 3u + 1u, next_step, &n2, &n3);
    gauss_pair(t * 3u + 2u, next_step, &n4, &nx);  // nx discarded
    u0 = clampf(u0 + NOISE_STD_C * n0, -1.0f, 1.0f);
    u1 = clampf(u1 + NOISE_STD_C * n1, -1.0f, 1.0f);
    u2 = clampf(u2 + NOISE_STD_C * n2, -1.0f, 1.0f);
    u3 = clampf(u3 + NOISE_STD_C * n3, -1.0f, 1.0f);
    u4 = clampf(u4 + NOISE_STD_C * n4, -1.0f, 1.0f);
  }

  float* O = xout + b * 5 * HW;
  O[0*HW+p] = u0;
  O[1*HW+p] = u1;
  O[2*HW+p] = u2;
  O[3*HW+p] = u3;
  O[4*HW+p] = u4;
}

// ---------------- energy terms (Q is a compile-time template parameter) ----------------
__device__ __forceinline__ float smooth_term(const float* __restrict__ F, int h, int w, float sc) {
  int p = h * WDIM + w;
  float f0 = F[p] * sc, f1 = F[HW + p] * sc;
  float dx0 = 0.f, dx1 = 0.f, dy0 = 0.f, dy1 = 0.f;
  if (w < WDIM - 1) { dx0 = F[p + 1] * sc - f0;      dx1 = F[HW + p + 1] * sc - f1; }
  if (h < HDIM - 1) { dy0 = F[p + WDIM] * sc - f0;   dy1 = F[HW + p + WDIM] * sc - f1; }
  return sqrtf(dx0 * dx0 + dx1 * dx1 + dy0 * dy0 + dy1 * dy1 + EPS_C);
}

template <int Q>
__device__ __forceinline__ float term_value(int b, int i,
                                            const float* __restrict__ t1,
                                            const float* __restrict__ in1,
                                            const float* __restrict__ in2,
                                            const float* __restrict__ xc) {
  int h = i >> 9, w = i & (WDIM - 1);
  if (Q == 0) {        // pos data:   img1=input1, img2=input2
    float s = EPS_C;
    #pragma unroll
    for (int c = 0; c < 3; ++c) {
      float d = in1[(b * 3 + c) * HW + i] - in2[(b * 3 + c) * HW + i];
      s += d * d;
    }
    return sqrtf(s);
  } else if (Q == 1) { // pos smooth: flow_pos*80 == target1
    return smooth_term(t1 + b * 2 * HW, h, w, 1.0f);
  } else if (Q == 2) { // neg data:   img1=(x+1)/2
    float s = EPS_C;
    #pragma unroll
    for (int c = 0; c < 3; ++c) {
      float d = 0.5f * (xc[(b * 5 + 2 + c) * HW + i] + 1.0f) - in2[(b * 3 + c) * HW + i];
      s += d * d;
    }
    return sqrtf(s);
  }
  return smooth_term(xc + b * 5 * HW, h, w, MAX_FLOW_C);  // neg smooth: flow = x*80
}

// Block-partial reduction; inner accumulation via V_WMMA_F32_16X16X4_F32 with
// an all-ones A matrix: D[m][n] = sum_k B[k][n]  -> 64 fp32 adds / instruction.
// Loop bound is read-first-laned to an SGPR so the loop closes with a scalar
// branch (no exec churn) and EXEC is structurally all-1s at the WMMA.
template <int Q>
__global__ void energy_partial_kernel(const float* __restrict__ t1,
                                      const float* __restrict__ in1,
                                      const float* __restrict__ in2,
                                      const float* __restrict__ xc,
                                      float* __restrict__ partials) {
  const int b = blockIdx.y;
  const int lane = threadIdx.x & 31;
  const int wave = threadIdx.x >> 5;
  // wave-uniform start index, forced into an SGPR
  const int waveGlobal = __builtin_amdgcn_readfirstlane(blockIdx.x * 8 + wave);
  const int stride = EBLOCKS * 8 * 64;               // 12288; HW/stride == 16 exactly

  v8f c = {};
  v2f a; a[0] = 1.0f; a[1] = 1.0f;                   // all-ones A (16x4)
  // indices are provably < HW (max 196607), so the body is branch-free
  for (int base = waveGlobal * 64; base < HW; base += stride) {
    float v0 = term_value<Q>(b, base + lane,      t1, in1, in2, xc);
    float v1 = term_value<Q>(b, base + 32 + lane, t1, in1, in2, xc);
#if __has_builtin(__builtin_amdgcn_wmma_f32_16x16x4_f32)
    v2f bb; bb[0] = v0; bb[1] = v1;
    c = __builtin_amdgcn_wmma_f32_16x16x4_f32(false, a, false, bb, (short)0, c,
                                              false, false);
#else
    c[0] += v0 + v1;
#endif
  }
  float s = c[0];
#if __has_builtin(__builtin_amdgcn_wmma_f32_16x16x4_f32)
  // lanes 0..15 of c[0] hold the 16 column sums (rows identical); reduce 16 lanes
  s += __shfl_xor(s, 8);
  s += __shfl_xor(s, 4);
  s += __shfl_xor(s, 2);
  s += __shfl_xor(s, 1);
#else
  s += __shfl_xor(s, 16);
  s += __shfl_xor(s, 8);
  s += __shfl_xor(s, 4);
  s += __shfl_xor(s, 2);
  s += __shfl_xor(s, 1);
#endif
  __shared__ float red[8];
  if (lane == 0) red[wave] = s;
  __syncthreads();
  if (threadIdx.x == 0) {
    float tot = 0.0f;
    #pragma unroll
    for (int i = 0; i < 8; ++i) tot += red[i];
    partials[(Q * BATCH + b) * EBLOCKS + blockIdx.x] = tot;
  }
}

__global__ void energies_kernel(const float* __restrict__ partials,
                                const float* __restrict__ lw, float* __restrict__ out) {
  int t = threadIdx.x;
  if (t >= 16) return;
  int b = t & 7;
  int neg = t >> 3;
  int qd = neg ? 2 : 0, qs = neg ? 3 : 1;
  float w0 = __expf(lw[0]), w1 = __expf(lw[1]);
  float sd = 0.f, ss = 0.f;
  for (int i = 0; i < EBLOCKS; ++i) {
    sd += partials[(qd * BATCH + b) * EBLOCKS + i];
    ss += partials[(qs * BATCH + b) * EBLOCKS + i];
  }
  out[t] = (w0 * sd + w1 * ss) * (1.0f / (float)HW);
}

__global__ void outputs_kernel(const float* __restrict__ xc, float* __restrict__ out) {
  int t = blockIdx.x * blockDim.x + threadIdx.x;
  if (t >= NPIX) return;
  int b = t / HW;
  int p = t - b * HW;
  const float* X = xc + b * 5 * HW;
  float* of = out + 16 + b * 2 * HW;                     // flow_neg * 80
  of[p]        = X[p] * MAX_FLOW_C;
  of[HW + p]   = X[HW + p] * MAX_FLOW_C;
  float* oi = out + 16 + BATCH * 2 * HW + b * 3 * HW;    // (img1_neg + 1)/2
  oi[p]          = (X[2 * HW + p] + 1.0f) * 0.5f;
  oi[HW + p]     = (X[3 * HW + p] + 1.0f) * 0.5f;
  oi[2 * HW + p] = (X[4 * HW + p] + 1.0f) * 0.5f;
}

// ---------------- launch ----------------
extern "C" void kernel_launch(void* const* d_in, const int* in_sizes, int n_in,
                              void* d_out, int out_size, void* d_ws, size_t ws_size,
                              hipStream_t stream) {
  const float* target1 = (const float*)d_in[0];
  const float* input1  = (const float*)d_in[1];
  const float* input2  = (const float*)d_in[2];
  const float* initx   = (const float*)d_in[3];
  const float* lw      = (const float*)d_in[4];
  float* out = (float*)d_out;
  float* ws  = (float*)d_ws;

  float* xA       = ws;                 // NXTOT floats (ping)
  float* xB       = ws + NXTOT;         // NXTOT floats (pong)
  float* partials = ws + 2 * NXTOT;     // 4*8*EBLOCKS floats

  const int TB = 256;

  // x_noised_0 = clip(init + N_0)
  init_noise_kernel<<<(NXTOT / 2 + TB - 1) / TB, TB, 0, stream>>>(initx, xA);

  // 200 fused steps; step s also applies noise for step s+1 (except the last)
  float* cur = xA;
  float* nxt = xB;
  dim3 sg(8 * 96, BATCH);               // 64x4 tiles: 8 across, 96 down, per batch
  for (int s = 0; s < NSTEPS; ++s) {
    int add_noise = (s < NSTEPS - 1) ? 1 : 0;
    step_kernel<<<sg, TB, 0, stream>>>(cur, input2, lw, nxt, (unsigned)(s + 1), add_noise);
    float* tmp = cur; cur = nxt; nxt = tmp;
  }
  // final state is in `cur`

  dim3 eg(EBLOCKS, BATCH);
  energy_partial_kernel<0><<<eg, TB, 0, stream>>>(target1, input1, input2, cur, partials);
  energy_partial_kernel<1><<<eg, TB, 0, stream>>>(target1, input1, input2, cur, partials);
  energy_partial_kernel<2><<<eg, TB, 0, stream>>>(target1, input1, input2, cur, partials);
  energy_partial_kernel<3><<<eg, TB, 0, stream>>>(target1, input1, input2, cur, partials);
  energies_kernel<<<1, 32, 0, stream>>>(partials, lw, out);
  outputs_kernel<<<(NPIX + TB - 1) / TB, TB, 0, stream>>>(cur, out);
}